// HL1ACELoss_45414984188195
// MI455X (gfx1250) — compile-verified
//
#include <hip/hip_runtime.h>

// ---------------- problem constants ----------------
#define NB    15
#define BATCH 4
#define CH    4
#define NSP   (128*128*128)          // 2,097,152 voxels per (b,c) slice
#define TILE  1024                   // voxels per LDS tile
#define TPB   256                    // 8 wave32 per block
#define WAVES (TPB/32)
#define LOG2E 1.4426950408889634f
// boundaries = linspace(0, 1+eps, 16); bin = floor(p * 15/(1+eps)) clamped
#define BIN_SCALE (15.0f/(1.0f + 1.1920928955078125e-07f))
#define OPS_PER_TILE 5               // 4 logit channels + 1 label stream

// ---------------- CDNA5 async global->LDS path ----------------
#if defined(__has_builtin)
#if __has_builtin(__builtin_amdgcn_s_wait_asynccnt)
#define WAIT_ASYNC(N) __builtin_amdgcn_s_wait_asynccnt(N)
#endif
#endif
#ifndef WAIT_ASYNC
#define WAIT_ASYNC(N) asm volatile("s_wait_asynccnt %0" :: "n"(N) : "memory")
#endif

// Per-lane 16B async copy global -> LDS (ASYNCcnt-tracked, wave32).
// VDST VGPR carries the LDS byte address; low 32 bits of a generic LDS
// pointer are the LDS offset (aperture layout, ISA 10.2).
__device__ __forceinline__ void async_copy16(const void* __restrict__ gsrc,
                                             const void* ldst) {
  unsigned lds_addr = (unsigned)(unsigned long long)ldst;
  asm volatile("global_load_async_to_lds_b128 %0, %1, off"
               :: "v"(lds_addr), "v"(gsrc)
               : "memory");
}

// ---------------- kernel 1: streaming softmax + histogram ----------------
// hist layout: [B][C][NB][3] floats: {count, sum_p, sum_t}
__global__ __launch_bounds__(TPB) void ace_hist_kernel(
    const float* __restrict__ logits,   // (B, C, 128^3) f32
    const int*   __restrict__ labels,   // (B, 1, 128^3) int
    float*       __restrict__ hist) {
  __shared__ __align__(16) float buf[2][CH][TILE];      // 32 KB double buffer
  __shared__ __align__(16) int   lbuf[2][TILE];         // 8 KB label buffer
  __shared__ float lhist[WAVES][CH][NB][3];             // per-wave sub-hists

  const int tid  = threadIdx.x;
  const int wave = tid >> 5;                            // wave32 on gfx1250
  const int b    = blockIdx.y;

  const float* __restrict__ lb   = logits + (size_t)b * CH * NSP;
  const int*   __restrict__ labb = labels + (size_t)b * NSP;

  const int tilesPerBatch = NSP / TILE;                 // 2048
  const int tstep = gridDim.x;
  const int t0 = blockIdx.x;

  auto issue = [&](int t, int bi) {
    const size_t vb = (size_t)t * TILE + tid * 4;
    #pragma unroll
    for (int c = 0; c < CH; ++c)
      async_copy16(lb + (size_t)c * NSP + vb, &buf[bi][c][tid * 4]);
    async_copy16(labb + vb, &lbuf[bi][tid * 4]);
  };

  // kick off first tile, then zero private histograms while it streams in
  if (t0 < tilesPerBatch) issue(t0, 0);
  for (int i = tid; i < WAVES * CH * NB * 3; i += TPB)
    ((float*)lhist)[i] = 0.0f;

  int bi = 0;
  for (int t = t0; t < tilesPerBatch; t += tstep) {
    const int tn = t + tstep;
    if (tn < tilesPerBatch) {
      issue(tn, bi ^ 1);           // prefetch next tile into other buffer
      WAIT_ASYNC(OPS_PER_TILE);    // previous 5-op group retired (in-order)
    } else {
      WAIT_ASYNC(0);
    }
    __syncthreads();               // all waves' async portions landed

    // 4 contiguous voxels per thread -> ds_load_b128 per stream
    const float4 L0 = *(const float4*)&buf[bi][0][tid * 4];
    const float4 L1 = *(const float4*)&buf[bi][1][tid * 4];
    const float4 L2 = *(const float4*)&buf[bi][2][tid * 4];
    const float4 L3 = *(const float4*)&buf[bi][3][tid * 4];
    const int4   LB = *(const int4*)&lbuf[bi][tid * 4];

    const float l0a[4] = {L0.x, L0.y, L0.z, L0.w};
    const float l1a[4] = {L1.x, L1.y, L1.z, L1.w};
    const float l2a[4] = {L2.x, L2.y, L2.z, L2.w};
    const float l3a[4] = {L3.x, L3.y, L3.z, L3.w};
    const int   lba[4] = {LB.x, LB.y, LB.z, LB.w};

    #pragma unroll
    for (int j = 0; j < 4; ++j) {
      const float l0 = l0a[j], l1 = l1a[j], l2 = l2a[j], l3 = l3a[j];
      const float m  = fmaxf(fmaxf(l0, l1), fmaxf(l2, l3));
      const float e0 = __builtin_exp2f((l0 - m) * LOG2E);
      const float e1 = __builtin_exp2f((l1 - m) * LOG2E);
      const float e2 = __builtin_exp2f((l2 - m) * LOG2E);
      const float e3 = __builtin_exp2f((l3 - m) * LOG2E);
      const float inv = 1.0f / (e0 + e1 + e2 + e3);
      const int lab = lba[j];

      const float ps[CH] = {e0 * inv, e1 * inv, e2 * inv, e3 * inv};
      #pragma unroll
      for (int c = 0; c < CH; ++c) {
        const float p = ps[c];
        int bin = (int)(p * BIN_SCALE);
        bin = bin > (NB - 1) ? (NB - 1) : bin;
        float* cell = &lhist[wave][c][bin][0];
        atomicAdd(cell + 0, 1.0f);          // ds_add_f32 (no-return)
        atomicAdd(cell + 1, p);
        if (lab == c) atomicAdd(cell + 2, 1.0f);
      }
    }
    __syncthreads();               // done reading buf[bi] before it is reused
    bi ^= 1;
  }

  __syncthreads();
  // cross-wave reduce, then one global float atomic per cell
  for (int i = tid; i < CH * NB * 3; i += TPB) {
    float s = 0.0f;
    #pragma unroll
    for (int w = 0; w < WAVES; ++w) s += ((float*)lhist[w])[i];
    atomicAdd(&hist[(size_t)b * CH * NB * 3 + i], s);
  }
}

// ---------------- kernel 2: zero workspace histogram ----------------
__global__ void ace_zero_ws(float* __restrict__ ws, int n) {
  int i = blockIdx.x * blockDim.x + threadIdx.x;
  if (i < n) ws[i] = 0.0f;
}

// ---------------- kernel 3: epilogue -> scalar ----------------
__global__ __launch_bounds__(64) void ace_final_kernel(
    const float* __restrict__ hist, float* __restrict__ out) {
  __shared__ float acc[BATCH * CH];
  const int i = threadIdx.x;
  if (i < BATCH * CH) {
    const float* h = hist + i * NB * 3;
    float dsum = 0.0f, tsum = 0.0f;
    int nvalid = 0;
    for (int bn = 0; bn < NB; ++bn) {
      const float cnt = h[bn * 3 + 0];
      const float sp  = h[bn * 3 + 1];
      const float st  = h[bn * 3 + 2];
      tsum += st;
      if (cnt > 0.0f) {
        dsum += fabsf(sp - st) / cnt;     // |mean_p - mean_t|
        ++nvalid;
      }
    }
    const float ace = dsum / (float)(nvalid > 0 ? nvalid : 1);
    acc[i] = (tsum > 0.0f) ? ace : 0.0f;  // mask empty-target slices
  }
  __syncthreads();
  if (i == 0) {
    float s = 0.0f;
    for (int j = 0; j < BATCH * CH; ++j) s += acc[j];
    out[0] = s / (float)(BATCH * CH);
  }
}

// ---------------- host entry ----------------
extern "C" void kernel_launch(void* const* d_in, const int* in_sizes, int n_in,
                              void* d_out, int out_size, void* d_ws, size_t ws_size,
                              hipStream_t stream) {
  const float* logits = (const float*)d_in[0];
  const int*   labels = (const int*)d_in[1];
  float* hist = (float*)d_ws;               // B*C*NB*3 = 720 floats
  float* out  = (float*)d_out;

  const int nws = BATCH * CH * NB * 3;
  ace_zero_ws<<<dim3((nws + 255) / 256), dim3(256), 0, stream>>>(hist, nws);

  dim3 grid(256, BATCH);                    // 1024 blocks, 8 tiles each
  ace_hist_kernel<<<grid, dim3(TPB), 0, stream>>>(logits, labels, hist);

  ace_final_kernel<<<1, dim3(64), 0, stream>>>(hist, out);
}